// LocalFeatureAggregation_77232101916767
// MI455X (gfx1250) — compile-verified
//
#include <hip/hip_runtime.h>

typedef float v2f __attribute__((ext_vector_type(2)));
typedef float v8f __attribute__((ext_vector_type(8)));

#define B_  8
#define N_  4096
#define K_  20
#define CO_ 64
#define M_EDGES (B_*N_*K_)   // 655360 edges

// ---------------------------------------------------------------------------
// Kernel A: exact KNN (self excluded) + per-block partial feature moments.
// One block = 256 rows of one batch; whole batch pos tile (48KB) in LDS.
// ---------------------------------------------------------------------------
__global__ __launch_bounds__(256) void knn_kernel(const float* __restrict__ pos,
                                                  int* __restrict__ idxbuf,
                                                  float* __restrict__ partial) {
  __shared__ float sp[N_ * 3];     // 48 KB: all points of this batch
  __shared__ float red[14 * 8];    // per-wave partial moments
  const int b   = blockIdx.x >> 4;     // 16 blocks per batch
  const int blk = blockIdx.x & 15;
  const float* pb = pos + (size_t)b * N_ * 3;
  for (int i = threadIdx.x; i < N_ * 3; i += 256) sp[i] = pb[i];
  __syncthreads();

  const int row = blk * 256 + threadIdx.x;
  const float cx = sp[row*3+0], cy = sp[row*3+1], cz = sp[row*3+2];

  float dl[K_]; int il[K_];
#pragma unroll
  for (int j = 0; j < K_; ++j) { dl[j] = 3.4e38f; il[j] = 0; }

  for (int m = 0; m < N_; ++m) {
    float dx = sp[m*3+0] - cx;
    float dy = sp[m*3+1] - cy;
    float dz = sp[m*3+2] - cz;
    float d2 = dx*dx + dy*dy + dz*dz;
    if (m == row) d2 = 3.0e38f;          // exclude self (ref adds 1e10 to diag)
    if (d2 < dl[K_-1]) {                 // sorted-insert, fully unrolled (regs only)
      float dc = d2; int ic = m;
#pragma unroll
      for (int j = 0; j < K_; ++j) {
        bool s = dc < dl[j];
        float td = dl[j]; int ti = il[j];
        dl[j] = s ? dc : td;  il[j] = s ? ic : ti;
        dc    = s ? td : dc;  ic    = s ? ti : ic;
      }
    }
  }
  {
    int* o = idxbuf + ((size_t)b * N_ + row) * K_;
#pragma unroll
    for (int j = 0; j < K_; ++j) o[j] = il[j];
  }

  // feature moments of f = (dx,dy,dz,||d||): 4 sums + 10 upper-tri products
  float s[14];
#pragma unroll
  for (int t = 0; t < 14; ++t) s[t] = 0.f;
#pragma unroll
  for (int j = 0; j < K_; ++j) {
    int m = il[j];
    float f0 = sp[m*3+0] - cx;
    float f1 = sp[m*3+1] - cy;
    float f2 = sp[m*3+2] - cz;
    float f3 = sqrtf(dl[j]);
    s[0]+=f0; s[1]+=f1; s[2]+=f2; s[3]+=f3;
    s[4]+=f0*f0; s[5]+=f0*f1; s[6]+=f0*f2; s[7]+=f0*f3;
    s[8]+=f1*f1; s[9]+=f1*f2; s[10]+=f1*f3;
    s[11]+=f2*f2; s[12]+=f2*f3; s[13]+=f3*f3;
  }
  const int lane = threadIdx.x & 31;
  const int wv   = threadIdx.x >> 5;
#pragma unroll
  for (int t = 0; t < 14; ++t) {
#pragma unroll
    for (int o = 16; o >= 1; o >>= 1) s[t] += __shfl_xor(s[t], o, 32);
  }
  if (lane == 0) {
#pragma unroll
    for (int t = 0; t < 14; ++t) red[t*8 + wv] = s[t];
  }
  __syncthreads();
  if (threadIdx.x < 14) {
    float a = 0.f;
#pragma unroll
    for (int w2 = 0; w2 < 8; ++w2) a += red[threadIdx.x*8 + w2];
    partial[blockIdx.x * 14 + threadIdx.x] = a;   // deterministic two-stage
  }
}

// ---------------------------------------------------------------------------
// Kernel B: reduce partial moments; fold BatchNorm into W' and b'.
//   mu_o = W_o·E[f]+b_o ; var_o = W_o·Cov(f)·W_o^T (biased, like jnp.var)
// ---------------------------------------------------------------------------
__global__ __launch_bounds__(64) void fold_kernel(const float* __restrict__ partial,
                                                  const float* __restrict__ W,
                                                  const float* __restrict__ bias,
                                                  const float* __restrict__ gamma,
                                                  const float* __restrict__ beta,
                                                  float* __restrict__ Wf,
                                                  float* __restrict__ bf) {
  __shared__ float mom[14];
  if (threadIdx.x < 14) {
    float a = 0.f;
    for (int i = 0; i < B_*16; ++i) a += partial[i*14 + threadIdx.x];
    mom[threadIdx.x] = a;
  }
  __syncthreads();
  const int o = threadIdx.x;                 // channel 0..63
  const float invM = 1.0f / (float)M_EDGES;
  float fm[4];
#pragma unroll
  for (int c = 0; c < 4; ++c) fm[c] = mom[c] * invM;
  float cv[4][4];
  {
    int p = 4;
#pragma unroll
    for (int c = 0; c < 4; ++c)
#pragma unroll
      for (int d = c; d < 4; ++d) {
        float v = mom[p] * invM - fm[c]*fm[d];
        cv[c][d] = v; cv[d][c] = v; ++p;
      }
  }
  float w[4];
#pragma unroll
  for (int c = 0; c < 4; ++c) w[c] = W[o*4 + c];
  float mu = bias[o];
#pragma unroll
  for (int c = 0; c < 4; ++c) mu += w[c]*fm[c];
  float var = 0.f;
#pragma unroll
  for (int c = 0; c < 4; ++c)
#pragma unroll
    for (int d = 0; d < 4; ++d) var += w[c]*w[d]*cv[c][d];
  float a = gamma[o] * rsqrtf(var + 1e-5f);
#pragma unroll
  for (int c = 0; c < 4; ++c) Wf[o*4 + c] = a * w[c];
  bf[o] = beta[o] + a * (bias[o] - mu);
}

// ---------------------------------------------------------------------------
// Kernel C: fused edge-feature + (BN-folded) linear via WMMA f32 16x16x4
//           + ReLU + mean-pool over k.
// One wave = 4 nodes = 80 edges = 5 M-tiles; 4 N-tiles of 16 channels.
// A layout (16x4 f32): lanes 0-15 hold (K0,K1) of edge M=lane,
//                      lanes 16-31 hold (K2,K3) of edge M=lane-16.
// D layout: VGPR r, lanes 0-15 -> edge 16t+r ; lanes 16-31 -> edge 16t+8+r.
// ---------------------------------------------------------------------------
__global__ __launch_bounds__(128) void edge_mlp_kernel(const float* __restrict__ pos,
                                                       const int* __restrict__ idxbuf,
                                                       const float* __restrict__ Wf,
                                                       const float* __restrict__ bf,
                                                       float* __restrict__ out) {
  __shared__ int   sidx[4][80];
  __shared__ float scen[4][12];
  const int lane = threadIdx.x & 31;
  const int wv   = threadIdx.x >> 5;
  const int g      = blockIdx.x * 4 + wv;   // node-group of 4 nodes
  const int gnode0 = g * 4;
  const int b  = gnode0 >> 12;              // /4096
  const int n0 = gnode0 & (N_ - 1);
  const float* pb = pos + (size_t)b * N_ * 3;

  for (int e = lane; e < 80; e += 32)
    sidx[wv][e] = idxbuf[(size_t)(b * N_ + n0) * K_ + e];  // 4 consecutive nodes
  if (lane < 12)
    scen[wv][lane] = pb[(size_t)n0 * 3 + lane];
  __syncthreads();

  const bool hi  = (lane >= 16);
  const int  lme = lane & 15;

  // B operand tiles: B[K][N] = W'[channel N][feature K]
  v2f Bt[4]; float bv[4];
#pragma unroll
  for (int tn = 0; tn < 4; ++tn) {
    int ch = lme + 16 * tn;
    Bt[tn].x = Wf[ch*4 + (hi ? 2 : 0)];
    Bt[tn].y = Wf[ch*4 + (hi ? 3 : 1)];
    bv[tn]   = bf[ch];
  }

  float acc[16];                            // [node 0..3][ntile 0..3]
#pragma unroll
  for (int q = 0; q < 16; ++q) acc[q] = 0.f;

#pragma unroll
  for (int t = 0; t < 5; ++t) {
    const int edge = 16*t + lme;
    const int node = edge / 20;
    const int nb   = sidx[wv][edge];
    const float cx = scen[wv][node*3+0];
    const float cy = scen[wv][node*3+1];
    const float cz = scen[wv][node*3+2];
    const float dx = pb[(size_t)nb*3+0] - cx;
    const float dy = pb[(size_t)nb*3+1] - cy;
    const float dz = pb[(size_t)nb*3+2] - cz;
    const float ds = sqrtf(dx*dx + dy*dy + dz*dz);
    v2f A;
    A.x = hi ? dz : dx;                     // K2 : K0
    A.y = hi ? ds : dy;                     // K3 : K1
#pragma unroll
    for (int tn = 0; tn < 4; ++tn) {
      v8f C = { bv[tn],bv[tn],bv[tn],bv[tn],bv[tn],bv[tn],bv[tn],bv[tn] };
      C = __builtin_amdgcn_wmma_f32_16x16x4_f32(false, A, false, Bt[tn],
                                                (short)0, C, false, false);
#pragma unroll
      for (int r = 0; r < 8; ++r) {
        float v = fmaxf((float)C[r], 0.0f);     // relu (leaky after relu = noop)
        const int nA = (16*t + r)     / 20;     // node of edge in low half
        const int nB = (16*t + r + 8) / 20;     // node of edge in high half
        if (nA == nB) {
          acc[nA*4 + tn] += v;
        } else {
          acc[nA*4 + tn] += hi ? 0.0f : v;
          acc[nB*4 + tn] += hi ? v : 0.0f;
        }
      }
    }
  }
  // combine lane halves (same channel, disjoint edges)
#pragma unroll
  for (int q = 0; q < 16; ++q) acc[q] += __shfl_xor(acc[q], 16, 32);

  if (!hi) {
    float* ob = out + ((size_t)b * N_ + n0) * CO_;
#pragma unroll
    for (int nd = 0; nd < 4; ++nd)
#pragma unroll
      for (int tn = 0; tn < 4; ++tn)
        ob[nd*CO_ + lme + 16*tn] = acc[nd*4 + tn] * (1.0f / K_);
  }
}

// ---------------------------------------------------------------------------
extern "C" void kernel_launch(void* const* d_in, const int* in_sizes, int n_in,
                              void* d_out, int out_size, void* d_ws, size_t ws_size,
                              hipStream_t stream) {
  // inputs: x[8,4096,3] (unused), pos[8,4096,3], W[64,4], b[64], gamma[64], beta[64]
  const float* pos   = (const float*)d_in[1];
  const float* W     = (const float*)d_in[2];
  const float* bias  = (const float*)d_in[3];
  const float* gamma = (const float*)d_in[4];
  const float* beta  = (const float*)d_in[5];
  float* out = (float*)d_out;

  int*   idxbuf  = (int*)d_ws;                      // B*N*K ints (2.62 MB)
  float* partial = (float*)(idxbuf + M_EDGES);      // 128*14 floats
  float* Wf      = partial + (B_*16) * 14;          // 64*4 folded weights
  float* bf      = Wf + CO_ * 4;                    // 64 folded biases

  hipLaunchKernelGGL(knn_kernel, dim3(B_ * 16), dim3(256), 0, stream,
                     pos, idxbuf, partial);
  hipLaunchKernelGGL(fold_kernel, dim3(1), dim3(64), 0, stream,
                     partial, W, bias, gamma, beta, Wf, bf);
  hipLaunchKernelGGL(edge_mlp_kernel, dim3((B_ * N_) / 16), dim3(128), 0, stream,
                     pos, idxbuf, Wf, bf, out);
}